// PointSparseAttn_75694503625221
// MI455X (gfx1250) — compile-verified
//
#include <hip/hip_runtime.h>
#include <math.h>

#define B_   4
#define N_   2048
#define C_   256
#define DFF_ 1024
#define NH_  4
#define KNN_ 16
#define DH_  64

typedef float v2f __attribute__((ext_vector_type(2)));
typedef float v8f __attribute__((ext_vector_type(8)));

enum { F_BIAS = 1, F_GELU = 2, F_RES = 4, F_TRANS = 8 };

// ---- CDNA5 async global->LDS copy (ASYNCcnt path), with sync fallback ----
#if defined(__has_builtin)
#  if __has_builtin(__builtin_amdgcn_global_load_async_to_lds_b32)
#    define HAS_ASYNC_LDS 1
#  endif
#endif
#ifndef HAS_ASYNC_LDS
#  define HAS_ASYNC_LDS 0
#endif

typedef __attribute__((address_space(1))) int gint_t;   // global
typedef __attribute__((address_space(3))) int lint_t;   // LDS

__device__ __forceinline__ void async_copy_b32(const float* g, float* l) {
#if HAS_ASYNC_LDS
    __builtin_amdgcn_global_load_async_to_lds_b32((gint_t*)g, (lint_t*)l, 0, 0);
#else
    *l = *g;   // synchronous fallback (ds_store path)
#endif
}

template <int CNT>
__device__ __forceinline__ void wait_async() {
#if HAS_ASYNC_LDS
#  if __has_builtin(__builtin_amdgcn_s_wait_asynccnt)
    __builtin_amdgcn_s_wait_asynccnt(CNT);
#  else
    asm volatile("s_wait_asynccnt %0" :: "i"(CNT) : "memory");
#  endif
#endif
}

__device__ __forceinline__ float gelu_exact(float x) {
    return 0.5f * x * (1.0f + erff(x * 0.70710678118654752f));
}

// D[m,col] = epilogue( sum_k A[m*sAm + k*sAk] * W[col*K + k] )
// All of M, Nc, K are multiples of 32 in this workload (no bounds checks).
// Double-buffered LDS tiles; tile k+1 streamed by the async DMA path while
// tile k feeds v_wmma_f32_16x16x4_f32.
__global__ __launch_bounds__(128)
void gemm_wmma(const float* __restrict__ A, long sAm, long sAk, long aBatch,
               const float* __restrict__ W,
               const float* __restrict__ bias,
               const float* __restrict__ resid,
               float* __restrict__ D, long dBatch,
               int M, int Nc, int K, int flags, int ldn)
{
    A += (size_t)blockIdx.z * aBatch;
    D += (size_t)blockIdx.z * dBatch;

    __shared__ float As[2][32][33];
    __shared__ float Bs[2][32][33];

    const int tid  = threadIdx.x;
    const int lane = tid & 31;
    const int wid  = tid >> 5;
    const int mo   = (wid >> 1) * 16;   // wave's M offset within 32x32 tile
    const int no   = (wid & 1) * 16;    // wave's N offset
    const int m0   = blockIdx.y * 32;
    const int n0   = blockIdx.x * 32;
    const int l    = lane & 15;
    const int half = lane >> 4;

    // issue one 32x32 A tile + 32x32 B tile into buffer `buf` (16 b32 ops/thread)
    auto issue = [&](int k0, int buf) {
        #pragma unroll
        for (int e = 0; e < 8; ++e) {
            int lin = tid + e * 128;
            int rr  = lin >> 5;
            int kc  = lin & 31;
            async_copy_b32(&A[(size_t)(m0 + rr) * sAm + (size_t)(k0 + kc) * sAk],
                           &As[buf][rr][kc]);
            async_copy_b32(&W[(size_t)(n0 + rr) * K + (k0 + kc)],
                           &Bs[buf][rr][kc]);
        }
    };

    v8f acc = {0.f, 0.f, 0.f, 0.f, 0.f, 0.f, 0.f, 0.f};

    const int nkt = K >> 5;
    issue(0, 0);
    for (int kt = 0; kt < nkt; ++kt) {
        const int buf = kt & 1;
        if (kt + 1 < nkt) {
            issue((kt + 1) << 5, buf ^ 1);   // prefetch next tile into other buffer
            wait_async<16>();                // in-order completion: current tile ready
        } else {
            wait_async<0>();
        }
        __syncthreads();
        #pragma unroll
        for (int kk = 0; kk < 8; ++kk) {
            // A 16x4 f32 layout: lanes 0-15 hold K={0,1}, lanes 16-31 K={2,3}
            v2f a, b;
            a.x = As[buf][mo + l][kk * 4 + 2 * half + 0];
            a.y = As[buf][mo + l][kk * 4 + 2 * half + 1];
            b.x = Bs[buf][no + l][kk * 4 + 2 * half + 0];
            b.y = Bs[buf][no + l][kk * 4 + 2 * half + 1];
            acc = __builtin_amdgcn_wmma_f32_16x16x4_f32(
                false, a, false, b, (short)0, acc, false, false);
        }
        __syncthreads();   // all waves done reading `buf` before it is refilled
    }

    // C/D layout: VGPR j -> (M=j, lanes 0-15) / (M=j+8, lanes 16-31), N=lane&15
    #pragma unroll
    for (int j = 0; j < 8; ++j) {
        int m   = m0 + mo + j + 8 * half;
        int col = n0 + no + l;
        float val = acc[j];
        if (flags & F_BIAS) val += bias[col];
        if (flags & F_GELU) val = gelu_exact(val);
        if (flags & F_RES)  val += resid[(size_t)m * Nc + col];
        if (flags & F_TRANS) {
            int bb = m / ldn, nn = m - bb * ldn;
            D[((size_t)bb * Nc + col) * ldn + nn] = val;
        } else {
            D[(size_t)m * Nc + col] = val;
        }
    }
}

__global__ __launch_bounds__(256)
void ln_kernel(const float* __restrict__ x, const float* __restrict__ g,
               const float* __restrict__ bt, float* __restrict__ y)
{
    __shared__ float red[256];
    const int row = blockIdx.x, tid = threadIdx.x;
    float v = x[(size_t)row * C_ + tid];
    red[tid] = v;
    __syncthreads();
    for (int s = 128; s > 0; s >>= 1) {
        if (tid < s) red[tid] += red[tid + s];
        __syncthreads();
    }
    float mean = red[0] * (1.0f / C_);
    __syncthreads();
    float d = v - mean;
    red[tid] = d * d;
    __syncthreads();
    for (int s = 128; s > 0; s >>= 1) {
        if (tid < s) red[tid] += red[tid + s];
        __syncthreads();
    }
    float var = red[0] * (1.0f / C_);
    y[(size_t)row * C_ + tid] = d * rsqrtf(var + 1e-5f) * g[tid] + bt[tid];
}

__global__ __launch_bounds__(64)
void knn_kernel(const float* __restrict__ pos, int* __restrict__ idx)
{
    int t = blockIdx.x * blockDim.x + threadIdx.x;
    if (t >= B_ * N_) return;
    int b = t / N_, qi = t - b * N_;
    const float* px = pos + (size_t)b * 3 * N_;
    float qx = px[qi], qy = px[N_ + qi], qz = px[2 * N_ + qi];
    float sqq = qx * qx + qy * qy + qz * qz;
    float best[KNN_];
    int   bidx[KNN_];
    #pragma unroll
    for (int i = 0; i < KNN_; ++i) { best[i] = 3.4e38f; bidx[i] = 0; }
    for (int j = 0; j < N_; ++j) {
        float jx = px[j], jy = px[N_ + j], jz = px[2 * N_ + j];
        float sqj = jx * jx + jy * jy + jz * jz;
        float d = sqq + sqj - 2.0f * (qx * jx + qy * jy + qz * jz);
        if (d < best[KNN_ - 1]) {
            int p = KNN_ - 1;
            while (p > 0 && d < best[p - 1]) {
                best[p] = best[p - 1]; bidx[p] = bidx[p - 1]; --p;
            }
            best[p] = d; bidx[p] = j;   // stable: ties keep earlier index first
        }
    }
    #pragma unroll
    for (int i = 0; i < KNN_; ++i) idx[(size_t)t * KNN_ + i] = bidx[i];
}

__global__ __launch_bounds__(128)
void relpos_kernel(const float* __restrict__ pos, const int* __restrict__ idx,
                   const float* __restrict__ Wr1, const float* __restrict__ br1,
                   const float* __restrict__ Wr2, const float* __restrict__ br2,
                   float* __restrict__ biasv)
{
    int t = blockIdx.x * blockDim.x + threadIdx.x;
    if (t >= B_ * N_ * KNN_) return;
    int bq = t / KNN_;
    int b = bq / N_, qi = bq - b * N_;
    int j = idx[t];
    const float* px = pos + (size_t)b * 3 * N_;
    float rx = px[qi]          - px[j];
    float ry = px[N_ + qi]     - px[N_ + j];
    float rz = px[2 * N_ + qi] - px[2 * N_ + j];
    float acc = 0.f;
    for (int o = 0; o < C_ / 4; ++o) {
        float h = Wr1[o * 3 + 0] * rx + Wr1[o * 3 + 1] * ry +
                  Wr1[o * 3 + 2] * rz + br1[o];
        acc += Wr2[o] * gelu_exact(h);
    }
    biasv[t] = acc + br2[0];
}

// one block per (b,q); wave w = head w; softmax over the 16 KNN entries only
// (bias is -inf elsewhere -> zero attention outside neighbors)
__global__ __launch_bounds__(128)
void sparse_attn_kernel(const float* __restrict__ q, const float* __restrict__ k,
                        const float* __restrict__ v, const int* __restrict__ idx,
                        const float* __restrict__ biasv, float* __restrict__ o)
{
    __shared__ float attn_s[NH_][KNN_];
    const int bq   = blockIdx.x;
    const int b    = bq / N_;
    const int tid  = threadIdx.x;
    const int lane = tid & 31;
    const int h    = tid >> 5;
    const int ks   = lane & 15;
    const int half = lane >> 4;

    const float* qp = q + (size_t)bq * C_ + h * DH_;
    int j = idx[(size_t)bq * KNN_ + ks];
    const float* kp = k + ((size_t)b * N_ + j) * C_ + h * DH_;
    float part = 0.f;
    #pragma unroll
    for (int d = 0; d < 32; ++d) part += qp[half * 32 + d] * kp[half * 32 + d];
    part += __shfl_xor(part, 16, 32);                    // full dot in both halves
    float score = part * 0.125f + biasv[(size_t)bq * KNN_ + ks];  // 1/sqrt(64)
    float mx = score;
    #pragma unroll
    for (int off = 8; off >= 1; off >>= 1) mx = fmaxf(mx, __shfl_xor(mx, off, 32));
    float e = expf(score - mx);
    float s = e;
    #pragma unroll
    for (int off = 8; off >= 1; off >>= 1) s += __shfl_xor(s, off, 32);
    if (lane < 16) attn_s[h][lane] = e / s;
    __syncthreads();

    float a0 = 0.f, a1 = 0.f;
    for (int kk = 0; kk < KNN_; ++kk) {
        int jj = idx[(size_t)bq * KNN_ + kk];
        const float* vp = v + ((size_t)b * N_ + jj) * C_ + h * DH_;
        float w = attn_s[h][kk];
        a0 += w * vp[lane];
        a1 += w * vp[lane + 32];
    }
    o[(size_t)bq * C_ + h * DH_ + lane]      = a0;
    o[(size_t)bq * C_ + h * DH_ + lane + 32] = a1;
}

extern "C" void kernel_launch(void* const* d_in, const int* in_sizes, int n_in,
                              void* d_out, int out_size, void* d_ws, size_t ws_size,
                              hipStream_t stream)
{
    (void)in_sizes; (void)n_in; (void)out_size; (void)ws_size;
    const float* src1 = (const float*)d_in[0];
    const float* src2 = (const float*)d_in[1];
    const float* pos  = (const float*)d_in[2];
    const float* Wp   = (const float*)d_in[3];
    const float* bp   = (const float*)d_in[4];
    const float* Wr1  = (const float*)d_in[5];
    const float* br1  = (const float*)d_in[6];
    const float* Wr2  = (const float*)d_in[7];
    const float* br2  = (const float*)d_in[8];
    const float* Wqkv = (const float*)d_in[9];
    const float* bqkv = (const float*)d_in[10];
    const float* Wo   = (const float*)d_in[11];
    const float* bo   = (const float*)d_in[12];
    const float* g13  = (const float*)d_in[13];
    const float* b13  = (const float*)d_in[14];
    const float* g12  = (const float*)d_in[15];
    const float* b12  = (const float*)d_in[16];
    const float* W1   = (const float*)d_in[17];
    const float* b1   = (const float*)d_in[18];
    const float* W2   = (const float*)d_in[19];
    const float* b2   = (const float*)d_in[20];

    float* ws = (float*)d_ws;
    const size_t SZ = (size_t)B_ * N_ * C_;    // 2,097,152 floats
    float* s1  = ws + 0 * SZ;
    float* s2  = ws + 1 * SZ;
    float* x1  = ws + 2 * SZ;
    float* x2  = ws + 3 * SZ;
    float* qb  = ws + 4 * SZ;
    float* kb  = ws + 5 * SZ;
    float* vb  = ws + 6 * SZ;
    float* ob  = ws + 7 * SZ;
    float* yb  = ws + 8 * SZ;
    float* x1n = ws + 9 * SZ;
    float* hb  = ws + 10 * SZ;                  // B*N*DFF = 4*SZ -> [10SZ,14SZ)
    int*   idx = (int*)(ws + 14 * SZ);          // B*N*16 ints
    float* bv  = ws + 14 * SZ + (size_t)B_ * N_ * KNN_;

    dim3 blk(128);

    // input_proj: s = Wp @ src  (A strided over (B,C,N), batched over B)
    gemm_wmma<<<dim3(C_ / 32, N_ / 32, B_), blk, 0, stream>>>(
        src1, 1, N_, (long)C_ * N_, Wp, bp, nullptr, s1, (long)N_ * C_,
        N_, C_, C_, F_BIAS, 0);
    gemm_wmma<<<dim3(C_ / 32, N_ / 32, B_), blk, 0, stream>>>(
        src2, 1, N_, (long)C_ * N_, Wp, bp, nullptr, s2, (long)N_ * C_,
        N_, C_, C_, F_BIAS, 0);

    ln_kernel<<<B_ * N_, 256, 0, stream>>>(s1, g13, b13, x1);
    ln_kernel<<<B_ * N_, 256, 0, stream>>>(s2, g13, b13, x2);

    // QKV projections
    gemm_wmma<<<dim3(C_ / 32, (B_ * N_) / 32, 1), blk, 0, stream>>>(
        x1, C_, 1, 0, Wqkv, bqkv, nullptr, qb, 0,
        B_ * N_, C_, C_, F_BIAS, 0);
    gemm_wmma<<<dim3(C_ / 32, (B_ * N_) / 32, 1), blk, 0, stream>>>(
        x2, C_, 1, 0, Wqkv + (size_t)C_ * C_, bqkv + C_, nullptr, kb, 0,
        B_ * N_, C_, C_, F_BIAS, 0);
    gemm_wmma<<<dim3(C_ / 32, (B_ * N_) / 32, 1), blk, 0, stream>>>(
        x2, C_, 1, 0, Wqkv + 2 * (size_t)C_ * C_, bqkv + 2 * C_, nullptr, vb, 0,
        B_ * N_, C_, C_, F_BIAS, 0);

    knn_kernel<<<(B_ * N_ + 63) / 64, 64, 0, stream>>>(pos, idx);
    relpos_kernel<<<(B_ * N_ * KNN_ + 127) / 128, 128, 0, stream>>>(
        pos, idx, Wr1, br1, Wr2, br2, bv);

    sparse_attn_kernel<<<B_ * N_, 128, 0, stream>>>(qb, kb, vb, idx, bv, ob);

    // out-proj + residual (post-LN x1)
    gemm_wmma<<<dim3(C_ / 32, (B_ * N_) / 32, 1), blk, 0, stream>>>(
        ob, C_, 1, 0, Wo, bo, x1, yb, 0, B_ * N_, C_, C_, F_BIAS | F_RES, 0);

    ln_kernel<<<B_ * N_, 256, 0, stream>>>(yb, g12, b12, x1n);

    // FFN1 (GELU) and FFN2 (+residual, transposed store to (B,C,N) output)
    gemm_wmma<<<dim3(DFF_ / 32, (B_ * N_) / 32, 1), blk, 0, stream>>>(
        x1n, C_, 1, 0, W1, b1, nullptr, hb, 0,
        B_ * N_, DFF_, C_, F_BIAS | F_GELU, 0);
    gemm_wmma<<<dim3(C_ / 32, (B_ * N_) / 32, 1), blk, 0, stream>>>(
        hb, DFF_, 1, 0, W2, b2, x1n, (float*)d_out, 0,
        B_ * N_, C_, DFF_, F_BIAS | F_RES | F_TRANS, N_);
}